// knnLoss_47038481826617
// MI455X (gfx1250) — compile-verified
//
#include <hip/hip_runtime.h>
#include <math.h>

typedef __attribute__((ext_vector_type(2))) float v2f;
typedef __attribute__((ext_vector_type(8))) float v8f;

#define N_PTS 8192
#define CHUNK 2048
#define THREADS 256
#define WAVES_PER_BLOCK 8                 // 8 wave32s, one 16-row source tile each
#define ROWS_PER_BLOCK (WAVES_PER_BLOCK*16)
#define KNN 3

// --- NaN-aware min/max without LLVM's canonicalize padding (asm is opaque) ---
__device__ __forceinline__ float vmin2(float a, float b) {
    float d;
    asm("v_min_num_f32 %0, %1, %2" : "=v"(d) : "v"(a), "v"(b));
    return d;
}
__device__ __forceinline__ float vmaxmin(float a, float b, float c) {
    // min(max(a,b), c)
    float d;
    asm("v_maxmin_num_f32 %0, %1, %2, %3" : "=v"(d) : "v"(a), "v"(b), "v"(c));
    return d;
}
__device__ __forceinline__ float vmax3(float a, float b, float c) {
    float d;
    asm("v_max3_num_f32 %0, %1, %2, %3" : "=v"(d) : "v"(a), "v"(b), "v"(c));
    return d;
}

// branchless sorted-insert of v into (m0<=m1<=m2): 4 VALU ops, no canonicalizes
__device__ __forceinline__ void insert3(float v, float& m0, float& m1, float& m2) {
    float t = vmax3(v, m0, m1);       // max of all three candidates below m2
    m1 = vmaxmin(v, m0, m1);          // new middle
    m0 = vmin2(v, m0);                // new smallest
    m2 = vmin2(t, m2);                // new largest-of-3
}

__global__ __launch_bounds__(THREADS)
void knn_main(const float* __restrict__ src, const float* __restrict__ tgt,
              float* __restrict__ ws) {
    __shared__ float4 tgt_s[CHUNK];                     // 32 KB staged targets
    __shared__ float  red_s[WAVES_PER_BLOCK * 16 * 48]; // 24 KB top-3 merge buffer

    const int blocksPerBatch = N_PTS / ROWS_PER_BLOCK;  // 64
    const int b    = blockIdx.x / blocksPerBatch;
    const int tb   = blockIdx.x % blocksPerBatch;
    const int tid  = threadIdx.x;
    const int w    = tid >> 5;
    const int lane = tid & 31;
    const int half = lane >> 4;
    const int l15  = lane & 15;
    const int rowBase = tb * ROWS_PER_BLOCK + w * 16;

    const float* srcB = src + (size_t)b * N_PTS * 3;
    const float* tgtB = tgt + (size_t)b * N_PTS * 3;
    const float  INF  = __builtin_inff();

    // A operand (16x4 f32): row m = (sx, sy, sz, 1).
    // Lanes 0-15 hold K=0,1 ; lanes 16-31 hold K=2,3 (per ISA A layout).
    v2f Av;
    {
        int m = rowBase + l15;
        if (half == 0) { Av.x = srcB[m*3 + 0]; Av.y = srcB[m*3 + 1]; }
        else           { Av.x = srcB[m*3 + 2]; Av.y = 1.0f; }
    }

    // C operand: C[m][n] = s2[m]; VGPR r <-> row (r + 8*half) in this lane.
    union { v8f v; float f[8]; } Cm;
    #pragma unroll
    for (int r = 0; r < 8; ++r) {
        int m = rowBase + 8*half + r;
        float x = srcB[m*3+0], y = srcB[m*3+1], z = srcB[m*3+2];
        Cm.f[r] = x*x + y*y + z*z;
    }

    // Per-lane running top-3 (smallest d2) for 8 rows each.
    float m0[8], m1[8], m2[8];
    #pragma unroll
    for (int r = 0; r < 8; ++r) { m0[r] = INF; m1[r] = INF; m2[r] = INF; }

    for (int c0 = 0; c0 < N_PTS; c0 += CHUNK) {
        // Stage a chunk of targets into LDS as (-2x, -2y, -2z, t2|inf).
        for (int i = tid; i < CHUNK; i += THREADS) {
            int t = c0 + i;
            float x = tgtB[t*3+0], y = tgtB[t*3+1], z = tgtB[t*3+2];
            float t2 = x*x + y*y + z*z;
            bool valid = (x != 0.0f) | (y != 0.0f) | (z != 0.0f);
            tgt_s[i] = make_float4(-2.0f*x, -2.0f*y, -2.0f*z, valid ? t2 : INF);
        }
        __syncthreads();

        // Prefetch next chunk's global data while we grind on this one
        // (global_prefetch_b8; data is L2-resident after first pass anyway).
        if (c0 + CHUNK < N_PTS) {
            for (int i = tid; i < CHUNK; i += THREADS)
                __builtin_prefetch(&tgtB[(size_t)(c0 + CHUNK + i) * 3], 0, 0);
        }

        #pragma unroll 4
        for (int tile = 0; tile < CHUNK/16; ++tile) {
            // B operand (4x16 f32): lanes 0-15 supply rows K=0,1 ; lanes 16-31 rows K=2,3.
            const float* p = (const float*)&tgt_s[tile*16 + l15];
            v2f Bv;
            Bv.x = p[half*2 + 0];
            Bv.y = p[half*2 + 1];

            union { v8f v; float f[8]; } D;
            // D = A x B + C  ==  s2 + t2 - 2*dot  ==  squared distance tile
            D.v = __builtin_amdgcn_wmma_f32_16x16x4_f32(
                      false, Av, false, Bv, (short)0, Cm.v, false, false);

            #pragma unroll
            for (int r = 0; r < 8; ++r) insert3(D.f[r], m0[r], m1[r], m2[r]);
        }
        __syncthreads();
    }

    // Cross-lane merge: each row's global top-3 is among 16 lanes' local top-3 (48 cands).
    {
        int base = w * 16 * 48;
        #pragma unroll
        for (int r = 0; r < 8; ++r) {
            int rowM = 8*half + r;
            int o = base + rowM*48 + l15*3;
            red_s[o+0] = m0[r]; red_s[o+1] = m1[r]; red_s[o+2] = m2[r];
        }
    }
    __syncthreads();

    if (lane < 16) {
        int base = w*16*48 + lane*48;
        float a = INF, bm = INF, cm = INF;
        for (int i = 0; i < 48; ++i) insert3(red_s[base + i], a, bm, cm);
        float s = sqrtf(fmaxf(a,  0.0f))
                + sqrtf(fmaxf(bm, 0.0f))
                + sqrtf(fmaxf(cm, 0.0f));
        int m = rowBase + lane;
        float x = srcB[m*3+0], y = srcB[m*3+1], z = srcB[m*3+2];
        bool valid = (x != 0.0f) | (y != 0.0f) | (z != 0.0f);
        if (valid) {
            atomicAdd(&ws[b*2 + 0], s);     // sum of k distances over valid sources
            atomicAdd(&ws[b*2 + 1], 1.0f);  // valid source count
        }
    }
}

__global__ void knn_init(float* ws, int n) {
    int i = threadIdx.x;
    if (i < n) ws[i] = 0.0f;
}

__global__ void knn_fin(const float* __restrict__ ws, float* __restrict__ out, int B) {
    float acc = 0.0f;
    for (int b = 0; b < B; ++b) {
        float cnt = ws[b*2 + 1];
        float s   = ws[b*2 + 0];
        acc += (cnt > 0.0f) ? s / (cnt * (float)KNN) : 0.0f;
    }
    out[0] = acc / (float)B;
}

extern "C" void kernel_launch(void* const* d_in, const int* in_sizes, int n_in,
                              void* d_out, int out_size, void* d_ws, size_t ws_size,
                              hipStream_t stream) {
    const float* src = (const float*)d_in[0];
    const float* tgt = (const float*)d_in[1];
    float* out = (float*)d_out;
    float* ws  = (float*)d_ws;

    int B = in_sizes[0] / (N_PTS * 3);   // 4 for the reference setup

    knn_init<<<1, 32, 0, stream>>>(ws, 2*B);
    knn_main<<<dim3(B * (N_PTS / ROWS_PER_BLOCK)), THREADS, 0, stream>>>(src, tgt, ws);
    knn_fin<<<1, 1, 0, stream>>>(ws, out, B);
}